// GraphConvolution_37915971289734
// MI455X (gfx1250) — compile-verified
//
#include <hip/hip_runtime.h>

typedef __attribute__((ext_vector_type(2))) float v2f;
typedef __attribute__((ext_vector_type(8))) float v8f;

#define C_DIM 128   // C_in == C_out == 128

// ---------------------------------------------------------------------------
// Kernel 1: out[b,n,:] = bias[:]   (doubles as accumulator zero-init)
// float4-vectorized; 128 ch = 32 float4s per row -> bias index = i & 31
// ---------------------------------------------------------------------------
__global__ void __launch_bounds__(256)
gc_init_bias(float4* __restrict__ out, const float4* __restrict__ bias4, int total4) {
    int i = blockIdx.x * blockDim.x + threadIdx.x;
    if (i < total4) out[i] = bias4[i & 31];
}

// ---------------------------------------------------------------------------
// Kernel 2: z = x @ W^T  (rows = B*N, K = N = 128), fp32 WMMA 16x16x4.
// One block = 16-row tile; 8 waves each own one 16-col tile of the output.
// W (64KB) and the x row-tile (8KB) are staged in LDS.
// ---------------------------------------------------------------------------
__global__ void __launch_bounds__(256)
gc_gemm_xwT(const float* __restrict__ x, const float* __restrict__ W,
            float* __restrict__ z) {
    __shared__ float sW[C_DIM * C_DIM];   // 64 KB: W row-major (C_out x C_in)
    __shared__ float sA[16 * C_DIM];      //  8 KB: 16 rows of x

    const int tid  = threadIdx.x;
    const int wave = tid >> 5;            // 0..7 -> N-tile
    const int lane = tid & 31;
    const int row0 = blockIdx.x * 16;

    // cooperative LDS fill (float4)
    {
        const float4* Wv  = (const float4*)W;
        float4*       sWv = (float4*)sW;
        #pragma unroll
        for (int i = 0; i < (C_DIM * C_DIM / 4) / 256; ++i)
            sWv[tid + i * 256] = Wv[tid + i * 256];

        const float4* Xv  = (const float4*)(x + (size_t)row0 * C_DIM);
        float4*       sAv = (float4*)sA;
        #pragma unroll
        for (int i = 0; i < (16 * C_DIM / 4) / 256; ++i)
            sAv[tid + i * 256] = Xv[tid + i * 256];
    }
    __syncthreads();

    const int half = lane >> 4;           // 0: K pair {k0,k0+1}, 1: {k0+2,k0+3}
    const int l16  = lane & 15;           // M for A-frag, N for B-frag
    const int n0   = wave * 16;

    v8f acc = {};
    for (int k0 = 0; k0 < C_DIM; k0 += 4) {
        const int k = k0 + 2 * half;
        v2f a, b;
        a.x = sA[l16 * C_DIM + k];
        a.y = sA[l16 * C_DIM + k + 1];
        b.x = sW[(n0 + l16) * C_DIM + k];     // B[k][n] = W[n][k]
        b.y = sW[(n0 + l16) * C_DIM + k + 1];
        acc = __builtin_amdgcn_wmma_f32_16x16x4_f32(
            /*neg_a=*/false, a, /*neg_b=*/false, b,
            /*c_mod=*/(short)0, acc, /*reuse_a=*/false, /*reuse_b=*/false);
    }

    // D layout: acc[v] -> M = v + 8*half, N = l16
    float* zt = z + (size_t)row0 * C_DIM + n0;
    #pragma unroll
    for (int v = 0; v < 8; ++v) {
        const int m = v + 8 * half;
        zt[(size_t)m * C_DIM + l16] = acc[v];
    }
}

// ---------------------------------------------------------------------------
// Kernel 3: scatter-add  out[b, row[e], :] += val[e] * z[b, col[e], :]
// One wave per edge; lane covers 4 channels (float4 gather + 4 f32 atomics).
// z fits in the 192MB L2, so gathers are L2-resident.
// ---------------------------------------------------------------------------
__global__ void __launch_bounds__(256)
gc_scatter(const int* __restrict__ rowI, const int* __restrict__ colI,
           const float* __restrict__ val, const float* __restrict__ z,
           float* __restrict__ out, int E, int N, int B) {
    const int waveId = blockIdx.x * (blockDim.x >> 5) + (threadIdx.x >> 5);
    const int lane   = threadIdx.x & 31;
    if (waveId >= E) return;

    const int   r = rowI[waveId];
    const int   c = colI[waveId];
    const float v = val[waveId];
    const size_t NC = (size_t)N * C_DIM;
    const int ch = lane * 4;

    for (int b = 0; b < B; ++b) {
        const float4 zv = *(const float4*)(z + b * NC + (size_t)c * C_DIM + ch);
        float* o = out + b * NC + (size_t)r * C_DIM + ch;
        atomicAdd(o + 0, v * zv.x);
        atomicAdd(o + 1, v * zv.y);
        atomicAdd(o + 2, v * zv.z);
        atomicAdd(o + 3, v * zv.w);
    }
}

// ---------------------------------------------------------------------------
extern "C" void kernel_launch(void* const* d_in, const int* in_sizes, int n_in,
                              void* d_out, int out_size, void* d_ws, size_t ws_size,
                              hipStream_t stream) {
    const float* x     = (const float*)d_in[0];   // (B,N,128) f32
    const int*   adj   = (const int*)  d_in[1];   // (2,E) i32
    const float* adjv  = (const float*)d_in[2];   // (E,)  f32
    const float* W     = (const float*)d_in[3];   // (128,128) f32
    const float* bias  = (const float*)d_in[4];   // (128,) f32

    const int E    = in_sizes[2];
    const int B    = 4;
    const int N    = in_sizes[0] / (B * C_DIM);
    const int rows = B * N;                        // 200000 (multiple of 16)

    float* out = (float*)d_out;
    float* z   = (float*)d_ws;                     // B*N*128 f32 intermediate

    // 1) out = bias (broadcast)  -- also serves as accumulator init
    {
        const int total4 = out_size / 4;
        gc_init_bias<<<(total4 + 255) / 256, 256, 0, stream>>>(
            (float4*)out, (const float4*)bias, total4);
    }
    // 2) z = x @ W^T via fp32 WMMA
    {
        const int rowTiles = rows / 16;            // 12500
        gc_gemm_xwT<<<rowTiles, 256, 0, stream>>>(x, W, z);
    }
    // 3) edge scatter-add into out
    {
        const int wavesPerBlock = 256 / 32;
        const int blocks = (E + wavesPerBlock - 1) / wavesPerBlock;
        gc_scatter<<<blocks, 256, 0, stream>>>(adj, adj + E, adjv, z, out, E, N, B);
    }
}